// HGNN_11630771437844
// MI455X (gfx1250) — compile-verified
//
#include <hip/hip_runtime.h>

typedef __attribute__((ext_vector_type(2))) float v2f;
typedef __attribute__((ext_vector_type(8))) float v8f;

// ---------------------------------------------------------------------------
// WMMA GEMM: out[nrows,64] = op((A[nrows,K] (+bias_in)) @ W[K,64] (+bias_out)), op = optional ReLU
// One wave computes a 16x64 output slab using V_WMMA_F32_16X16X4_F32.
// f32 A-frag layout (ISA 7.12.2): lane m(=lane&15), VGPR j holds A[m][2*half+j]
// f32 B-frag: VGPR j holds B[k0+2*half+j][col=m] ; C VGPR r -> row r+8*half, col m.
// W[K,64] is staged into LDS with GLOBAL_LOAD_ASYNC_TO_LDS_B128 (ASYNCcnt path,
// bypasses VGPRs), then consumed with ds loads.
// ---------------------------------------------------------------------------
template <int K, bool RELU, bool ADD_IN_BIAS, bool ADD_OUT_BIAS>
__global__ __launch_bounds__(256) void gcn_gemm_n64(
    const float* __restrict__ A, const float* __restrict__ W,
    const float* __restrict__ bias_in, const float* __restrict__ bias_out,
    float* __restrict__ out, int nrows) {
  __shared__ __align__(16) float sW[K * 64];

  // ---- async stage W -> LDS (16B per transfer, per-lane addresses) ----
#pragma unroll 2
  for (int i = threadIdx.x; i < K * 16; i += 256) {
    unsigned lds_addr = (unsigned)(size_t)(&sW[i * 4]);  // wave-relative LDS byte addr
    unsigned voff = (unsigned)(i * 16);                  // byte offset from SGPR base
    asm volatile("global_load_async_to_lds_b128 %0, %1, %2"
                 :: "v"(lds_addr), "v"(voff), "s"(W)
                 : "memory");
  }
  asm volatile("s_wait_asynccnt 0x0" ::: "memory");
  __syncthreads();

  const int wave = threadIdx.x >> 5;
  const int lane = threadIdx.x & 31;
  const int half = lane >> 4;
  const int m = lane & 15;
  const int rt = blockIdx.x * 8 + wave;
  const int row0 = rt * 16;
  if (row0 >= nrows) return;  // wave-uniform: EXEC stays all-ones for WMMA

  int ar = row0 + m;
  if (ar >= nrows) ar = nrows - 1;  // clamp (harmless duplicate work)
  const float* __restrict__ arow = A + (size_t)ar * K;

  v8f acc[4] = {v8f{}, v8f{}, v8f{}, v8f{}};

#pragma unroll 4
  for (int kk = 0; kk < K / 4; ++kk) {
    const int k0 = kk * 4 + 2 * half;
    v2f a;
    float a0 = arow[k0];
    float a1 = arow[k0 + 1];
    if (ADD_IN_BIAS) {
      a0 += bias_in[k0];
      a1 += bias_in[k0 + 1];
    }
    a.x = a0;
    a.y = a1;
#pragma unroll
    for (int ct = 0; ct < 4; ++ct) {
      v2f b;
      b.x = sW[k0 * 64 + ct * 16 + m];
      b.y = sW[(k0 + 1) * 64 + ct * 16 + m];
      acc[ct] = __builtin_amdgcn_wmma_f32_16x16x4_f32(
          false, a, false, b, (short)0, acc[ct], false, false);
    }
  }

#pragma unroll
  for (int ct = 0; ct < 4; ++ct) {
    const float bo = ADD_OUT_BIAS ? bias_out[ct * 16 + m] : 0.0f;
#pragma unroll
    for (int r = 0; r < 8; ++r) {
      const int orow = row0 + r + 8 * half;
      float v = acc[ct][r] + bo;
      if (RELU) v = fmaxf(v, 0.0f);
      if (orow < nrows) out[(size_t)orow * 64 + ct * 16 + m] = v;
    }
  }
}

// ---------------------------------------------------------------------------
// Degree / normalization helpers
// ---------------------------------------------------------------------------
__global__ void deg_init_k(float* __restrict__ deg, int n) {
  int i = blockIdx.x * 256 + threadIdx.x;
  if (i < n) deg[i] = 1.0f;  // self-loop
}

__global__ void deg_count_k(const int* __restrict__ col, float* __restrict__ deg,
                            int ne) {
  int i = blockIdx.x * 256 + threadIdx.x;
  if (i < ne) atomicAdd(&deg[col[i]], 1.0f);
}

__global__ void dinv_k(const float* __restrict__ deg, float* __restrict__ dinv,
                       int n) {
  int i = blockIdx.x * 256 + threadIdx.x;
  if (i < n) {
    float d = deg[i];
    dinv[i] = (d > 0.0f) ? rsqrtf(d) : 0.0f;
  }
}

// agg[i,:] = dinv[i]^2 * hg[i,:]   (self-loop contribution; also zero-inits agg)
__global__ void agg_init_k(const float* __restrict__ hg,
                           const float* __restrict__ dinv,
                           float* __restrict__ agg, int n) {
  int idx = blockIdx.x * 256 + threadIdx.x;
  if (idx < n * 64) {
    int node = idx >> 6;
    float d = dinv[node];
    agg[idx] = hg[idx] * (d * d);
  }
}

// One wave32 per edge; lane covers channels (2*lane, 2*lane+1).
// hg/agg are 12.8MB each -> L2 (192MB) resident; atomics resolve at L2.
__global__ __launch_bounds__(256) void edge_scatter_k(
    const int* __restrict__ rows, const int* __restrict__ cols,
    const float* __restrict__ hg, const float* __restrict__ dinv,
    float* __restrict__ agg, int ne) {
  int e = (int)((blockIdx.x * 256 + threadIdx.x) >> 5);
  int lane = threadIdx.x & 31;
  if (e >= ne) return;
  int r = rows[e];
  int c = cols[e];
  float norm = dinv[r] * dinv[c];
  const float2* __restrict__ hrow =
      reinterpret_cast<const float2*>(hg + (size_t)r * 64);
  float2 v = hrow[lane];
  float* dst = agg + (size_t)c * 64 + lane * 2;
  atomicAdd(dst, norm * v.x);
  atomicAdd(dst + 1, norm * v.y);
}

// ---------------------------------------------------------------------------
extern "C" void kernel_launch(void* const* d_in, const int* in_sizes, int n_in,
                              void* d_out, int out_size, void* d_ws,
                              size_t ws_size, hipStream_t stream) {
  const float* x  = (const float*)d_in[0];
  const int*   ei = (const int*)d_in[1];  // [2, E]
  const float* W1 = (const float*)d_in[2];
  const float* b1 = (const float*)d_in[3];
  const float* Wg = (const float*)d_in[4];
  const float* bg = (const float*)d_in[5];
  const float* W2 = (const float*)d_in[6];
  const float* b2 = (const float*)d_in[7];
  float* out = (float*)d_out;

  const int N = in_sizes[0] / 128;
  const int E = in_sizes[1] / 2;
  const int* rows = ei;
  const int* cols = ei + E;

  char* ws = (char*)d_ws;
  const size_t featB = (size_t)N * 64 * sizeof(float);
  float* h    = (float*)(ws);
  float* hg   = (float*)(ws + featB);
  float* agg  = (float*)(ws + 2 * featB);
  float* deg  = (float*)(ws + 3 * featB);
  float* dinv = (float*)(ws + 3 * featB + (size_t)N * sizeof(float));

  const int tiles = (N + 15) / 16;
  const int gemmBlocks = (tiles + 7) / 8;
  const int nb = (N + 255) / 256;

  // 1) h = relu(x @ W1 + b1)
  gcn_gemm_n64<128, true, false, true>
      <<<gemmBlocks, 256, 0, stream>>>(x, W1, nullptr, b1, h, N);
  // 2) hg = h @ Wg
  gcn_gemm_n64<64, false, false, false>
      <<<gemmBlocks, 256, 0, stream>>>(h, Wg, nullptr, nullptr, hg, N);
  // 3) degrees (self-loop + in-edges on col), dinv = rsqrt(deg)
  deg_init_k<<<nb, 256, 0, stream>>>(deg, N);
  deg_count_k<<<(E + 255) / 256, 256, 0, stream>>>(cols, deg, E);
  dinv_k<<<nb, 256, 0, stream>>>(deg, dinv, N);
  // 4) agg = dinv^2 * hg (self-loop), then scatter-add normalized messages
  agg_init_k<<<((size_t)N * 64 + 255) / 256, 256, 0, stream>>>(hg, dinv, agg, N);
  edge_scatter_k<<<(E + 7) / 8, 256, 0, stream>>>(rows, cols, hg, dinv, agg, E);
  // 5) out = (agg + bg) @ W2 + b2
  gcn_gemm_n64<64, false, true, true>
      <<<gemmBlocks, 256, 0, stream>>>(agg, W2, bg, b2, out, N);
}